// DEMA_3607772528803
// MI455X (gfx1250) — compile-verified
//
#include <hip/hip_runtime.h>

// ---------------------------------------------------------------------------
// DEMA (Holt's linear trend) via chunked parallel scan + WMMA f32 16x16x4.
// state_t = M state_{t-1} + v x_t with constant M,v (alpha=0.3, beta=0.1).
// Chunk L=16: phase1 per-chunk contributions, phase2 state propagation,
// phase3 per-chunk outputs as a 16x20 @ 20x16 matmul on the WMMA pipe.
// ---------------------------------------------------------------------------

typedef __attribute__((ext_vector_type(2))) float v2f;
typedef __attribute__((ext_vector_type(8))) float v8f;

#define B_DIM 64
#define T_DIM 4096
#define D_DIM 64
#define NCH   (B_DIM * D_DIM)          // 4096 channels
#define L     16                        // chunk length
#define NCHK  (T_DIM / L)               // 256 chunks
#define XSTRIDE_B ((size_t)T_DIM * D_DIM)  // 262144

// workspace layout (floats)
#define WS_DS 0                         // d_c s-component  [256][4096]
#define WS_DB (NCHK * NCH)              // d_c b-component
#define WS_SS (2 * NCHK * NCH)          // chunk-start s    [256][4096]
#define WS_SB (3 * NCHK * NCH)          // chunk-start b

struct FTab {
    float Amat[16][20];   // row i: C[i][j]=c_{i-j} (j<=i), cols 16/17 = row0 of M^{i+1}, 18/19 = 0
    float pvs[16];        // (M^k v)[0]
    float pvb[16];        // (M^k v)[1]
    float m16[4];         // M^16 row-major
};

static constexpr FTab make_ftab() {
    const double a = 0.3, be = 0.1;
    const double M00 = 1.0 - a, M01 = 1.0 - a;
    const double M10 = -be * a, M11 = be * (1.0 - a) + (1.0 - be);
    double pvs[16] = {}, pvb[16] = {};
    double ps = a, pb = be * a;                 // M^0 v
    for (int k = 0; k < 16; ++k) {
        pvs[k] = ps; pvb[k] = pb;
        double ns = M00 * ps + M01 * pb;
        double nb = M10 * ps + M11 * pb;
        ps = ns; pb = nb;
    }
    FTab f{};
    for (int k = 0; k < 16; ++k) { f.pvs[k] = (float)pvs[k]; f.pvb[k] = (float)pvb[k]; }
    double p00 = M00, p01 = M01, p10 = M10, p11 = M11;   // M^1
    for (int i = 0; i < 16; ++i) {
        f.Amat[i][16] = (float)p00;   // first row of M^{i+1}
        f.Amat[i][17] = (float)p01;
        f.Amat[i][18] = 0.0f;
        f.Amat[i][19] = 0.0f;
        for (int j = 0; j < 16; ++j)
            f.Amat[i][j] = (j <= i) ? (float)pvs[i - j] : 0.0f;
        if (i == 15) {                // p == M^16 here
            f.m16[0] = (float)p00; f.m16[1] = (float)p01;
            f.m16[2] = (float)p10; f.m16[3] = (float)p11;
        }
        double q00 = M00 * p00 + M01 * p10, q01 = M00 * p01 + M01 * p11;
        double q10 = M10 * p00 + M11 * p10, q11 = M10 * p01 + M11 * p11;
        p00 = q00; p01 = q01; p10 = q10; p11 = q11;
    }
    return f;
}

__constant__ FTab cTab = make_ftab();

// ---------------------------------------------------------------------------
// Phase 1: d_c = sum_{j=0..15} (M^{15-j} v) * x[t0+1+j], for c = 0..254.
// ---------------------------------------------------------------------------
__global__ void dema_phase1(const float* __restrict__ x, float* __restrict__ ws) {
    int gid = blockIdx.x * blockDim.x + threadIdx.x;   // 255*4096 threads
    int c  = gid >> 12;
    int ch = gid & (NCH - 1);
    if (c >= NCHK - 1) return;
    int b = ch >> 6, d = ch & 63;
    const float* xp = x + (size_t)b * XSTRIDE_B + (size_t)(L * c + 1) * D_DIM + d;
    float ds = 0.0f, db = 0.0f;
#pragma unroll
    for (int j = 0; j < 16; ++j) {
        float xv = xp[(size_t)j * D_DIM];
        ds = fmaf(cTab.pvs[15 - j], xv, ds);
        db = fmaf(cTab.pvb[15 - j], xv, db);
    }
    ws[WS_DS + c * NCH + ch] = ds;
    ws[WS_DB + c * NCH + ch] = db;
}

// ---------------------------------------------------------------------------
// Phase 2: propagate chunk-start states: state_{c+1} = M^16 state_c + d_c.
// Also writes out[t=0] = x[t=0].
// ---------------------------------------------------------------------------
__global__ void dema_phase2(const float* __restrict__ x, float* __restrict__ out,
                            float* __restrict__ ws) {
    int ch = blockIdx.x * blockDim.x + threadIdx.x;    // 4096 threads
    int b = ch >> 6, d = ch & 63;
    const float* xb = x + (size_t)b * XSTRIDE_B + d;
    float s  = xb[0];
    float bb = xb[D_DIM] - s;
    out[(size_t)b * XSTRIDE_B + d] = s;                // t = 0 output
    const float m00 = cTab.m16[0], m01 = cTab.m16[1];
    const float m10 = cTab.m16[2], m11 = cTab.m16[3];
    for (int c = 0; c < NCHK; ++c) {
        ws[WS_SS + c * NCH + ch] = s;
        ws[WS_SB + c * NCH + ch] = bb;
        if (c < NCHK - 1) {
            float ns = fmaf(m00, s, fmaf(m01, bb, ws[WS_DS + c * NCH + ch]));
            float nb = fmaf(m10, s, fmaf(m11, bb, ws[WS_DB + c * NCH + ch]));
            s = ns; bb = nb;
        }
    }
}

// ---------------------------------------------------------------------------
// Phase 3: one wave per 16(time) x 16(channel) tile.
// D = A(16x20) * Xaug(20x16) via 5 chained V_WMMA_F32_16X16X4_F32.
// Xaug rows 0..15 = x rows, 16 = s-state, 17 = b-state, 18/19 = 0.
// ---------------------------------------------------------------------------
__global__ void dema_phase3(const float* __restrict__ x, float* __restrict__ out,
                            const float* __restrict__ ws) {
    int wave = (blockIdx.x * blockDim.x + threadIdx.x) >> 5;  // tile id 0..65535
    int lane = threadIdx.x & 31;
    int c  = wave & (NCHK - 1);
    int dt = (wave >> 8) & 3;
    int b  = wave >> 10;
    int half = lane >> 4;        // 0: low half of K, 1: high half of K
    int n    = lane & 15;        // matrix row (A) / column (B,D)
    int d0   = dt * 16;
    int t0   = c * L;            // outputs at t0+1 .. t0+16 (last chunk: ..t0+15)

    // A operands (constant tables): lane holds A[n][4kb + 2*half + {0,1}]
    v2f A0, A1, A2, A3, A4;
    {
        int ka0 = 0  + 2 * half;
        int ka1 = 4  + 2 * half;
        int ka2 = 8  + 2 * half;
        int ka3 = 12 + 2 * half;
        int ka4 = 16 + 2 * half;
        A0 = v2f{cTab.Amat[n][ka0], cTab.Amat[n][ka0 + 1]};
        A1 = v2f{cTab.Amat[n][ka1], cTab.Amat[n][ka1 + 1]};
        A2 = v2f{cTab.Amat[n][ka2], cTab.Amat[n][ka2 + 1]};
        A3 = v2f{cTab.Amat[n][ka3], cTab.Amat[n][ka3 + 1]};
        A4 = v2f{cTab.Amat[n][ka4], cTab.Amat[n][ka4 + 1]};
    }

    // B operands: lane holds Xaug[4kb + 2*half + {0,1}][n]
    const float* xb = x + (size_t)b * XSTRIDE_B + (size_t)(d0 + n);
    v2f Bv[5];
#pragma unroll
    for (int kb = 0; kb < 4; ++kb) {
        int r0 = 4 * kb + 2 * half;
        int ta = t0 + 1 + r0;
        int tb = ta + 1;
        float blo = (ta < T_DIM) ? xb[(size_t)ta * D_DIM] : 0.0f;
        float bhi = (tb < T_DIM) ? xb[(size_t)tb * D_DIM] : 0.0f;
        Bv[kb] = v2f{blo, bhi};
    }
    {   // kb = 4: rows 16 (s-state), 17 (b-state) for lanes 0-15; rows 18/19 = 0
        int chn = b * 64 + d0 + n;
        float slo = ws[WS_SS + c * NCH + chn];
        float shi = ws[WS_SB + c * NCH + chn];
        Bv[4] = v2f{half ? 0.0f : slo, half ? 0.0f : shi};
    }

    v8f acc = {};
    acc = __builtin_amdgcn_wmma_f32_16x16x4_f32(false, A0, false, Bv[0], (short)0, acc, false, false);
    acc = __builtin_amdgcn_wmma_f32_16x16x4_f32(false, A1, false, Bv[1], (short)0, acc, false, false);
    acc = __builtin_amdgcn_wmma_f32_16x16x4_f32(false, A2, false, Bv[2], (short)0, acc, false, false);
    acc = __builtin_amdgcn_wmma_f32_16x16x4_f32(false, A3, false, Bv[3], (short)0, acc, false, false);
    acc = __builtin_amdgcn_wmma_f32_16x16x4_f32(false, A4, false, Bv[4], (short)0, acc, false, false);

    // D layout: VGPR r, lanes 0-15 -> M=r, lanes 16-31 -> M=r+8, N=lane%16
#pragma unroll
    for (int r = 0; r < 8; ++r) {
        int it = r + 8 * half;          // 0..15 within chunk
        int t  = t0 + 1 + it;
        if (t < T_DIM)
            out[(size_t)b * XSTRIDE_B + (size_t)t * D_DIM + (d0 + n)] = acc[r];
    }
}

extern "C" void kernel_launch(void* const* d_in, const int* in_sizes, int n_in,
                              void* d_out, int out_size, void* d_ws, size_t ws_size,
                              hipStream_t stream) {
    (void)in_sizes; (void)n_in; (void)out_size; (void)ws_size;
    const float* x = (const float*)d_in[0];
    float* out = (float*)d_out;
    float* ws  = (float*)d_ws;   // needs 4 * 256 * 4096 floats = 16 MB

    // Phase 1: 255 chunks * 4096 channels, 256 threads/block
    dema_phase1<<<(NCHK - 1) * NCH / 256, 256, 0, stream>>>(x, ws);
    // Phase 2: 4096 channel threads
    dema_phase2<<<NCH / 256, 256, 0, stream>>>(x, out, ws);
    // Phase 3: 65536 tiles, 1 wave each, 8 waves/block
    dema_phase3<<<(B_DIM * 4 * NCHK) / 8, 256, 0, stream>>>(x, out, ws);
}